// EdgeDecoder_77343771066809
// MI455X (gfx1250) — compile-verified
//
#include <hip/hip_runtime.h>

typedef __attribute__((ext_vector_type(2))) float v2f;
typedef __attribute__((ext_vector_type(8))) float v8f;

#define HIDDEN    128
#define GROUP     16      // edges per tile (WMMA M)
#define ZT_STRIDE 260     // padded LDS row stride in floats (16B aligned, bank-conflict-free)
#define NWAVES    8       // 8 waves x 16 cols = 128 hidden

__global__ __launch_bounds__(256) void edge_decoder_kernel(
    const float* __restrict__ zs,       // [N_STUDENT, 128]
    const float* __restrict__ zc,       // [N_COURSE, 128]
    const long long* __restrict__ row,  // [E]
    const long long* __restrict__ col,  // [E]
    const float* __restrict__ W1,       // [256, 128] row-major
    const float* __restrict__ b1,       // [128]
    const float* __restrict__ W2,       // [128]
    const float* __restrict__ b2,       // [1]
    float* __restrict__ out,            // [E]
    int n_edges)
{
    __shared__ float zt[GROUP * ZT_STRIDE];     // gathered [16 edges x 256 feat] tile
    __shared__ float red[NWAVES][GROUP];        // cross-wave partial sums

    const int tid  = threadIdx.x;
    const int lane = tid & 31;
    const int w    = tid >> 5;          // wave id 0..7 -> owns hidden cols [16w, 16w+16)
    const int nloc = lane & 15;         // N (cols) / M (A rows) sub-lane
    const int kb   = (lane >> 4) * 2;   // K base within a 16x16x4 step per half-wave

    // ---- one-time per wave: register-resident B fragments of W1's 16-col slice ----
    const int n = w * 16 + nloc;
    float bf0[64], bf1[64];
    #pragma unroll
    for (int ks = 0; ks < 64; ++ks) {
        const int k = ks * 4 + kb;
        bf0[ks] = W1[k * HIDDEN + n];          // B[K=kb  ][N=n]
        bf1[ks] = W1[(k + 1) * HIDDEN + n];    // B[K=kb+1][N=n]
    }
    const float b1n = b1[n];
    const float w2n = W2[n];
    const float b2v = b2[0];

    const int ngroups = n_edges >> 4;

    for (int g = blockIdx.x; g < ngroups; g += gridDim.x) {
        // ---- cooperative gather: 16 threads per edge, 16 floats per thread ----
        {
            const int e    = tid >> 4;      // local edge 0..15
            const int part = tid & 15;      // which 16-float chunk of the 256
            const long long eidx = (long long)g * GROUP + e;
            const float* src;
            if (part < 8) {
                const long long r = row[eidx];
                src = zs + (size_t)r * HIDDEN + part * 16;
            } else {
                const long long c = col[eidx];
                src = zc + (size_t)c * HIDDEN + (part - 8) * 16;
            }
            float4 v0 = ((const float4*)src)[0];
            float4 v1 = ((const float4*)src)[1];
            float4 v2 = ((const float4*)src)[2];
            float4 v3 = ((const float4*)src)[3];
            float4* dst = (float4*)&zt[e * ZT_STRIDE + part * 16];
            dst[0] = v0; dst[1] = v1; dst[2] = v2; dst[3] = v3;

            // prefetch index cachelines for upcoming groups (global_prefetch_b8)
            if ((tid & 63) == 0) {
                __builtin_prefetch(&row[eidx + 8 * GROUP], 0, 1);
                __builtin_prefetch(&col[eidx + 8 * GROUP], 0, 1);
            }
        }
        __syncthreads();

        // ---- [16 edges x 16 cols] tile: 64 x V_WMMA_F32_16X16X4_F32 over K=256 ----
        v8f c = {};
        const float* arow = &zt[nloc * ZT_STRIDE + kb];   // A[M=nloc][K=ks*4+kb ..]
        #pragma unroll
        for (int ks = 0; ks < 64; ++ks) {
            v2f a = *(const v2f*)(arow + ks * 4);         // ds_load_b64, conflict-free
            v2f b; b.x = bf0[ks]; b.y = bf1[ks];
            c = __builtin_amdgcn_wmma_f32_16x16x4_f32(
                    /*neg_a=*/false, a, /*neg_b=*/false, b,
                    /*c_mod=*/(short)0, c, /*reuse_a=*/false, /*reuse_b=*/false);
        }

        // ---- h = relu(c + b1[n]); partial = h * W2[n]; reduce over 16 N-lanes ----
        float p[8];
        #pragma unroll
        for (int j = 0; j < 8; ++j) {
            float h = c[j] + b1n;
            h = h > 0.f ? h : 0.f;
            p[j] = h * w2n;
        }
        #pragma unroll
        for (int j = 0; j < 8; ++j) {
            #pragma unroll
            for (int m = 1; m < 16; m <<= 1)
                p[j] += __shfl_xor(p[j], m, 32);
        }
        // C layout: lanes 0-15 hold rows M=j, lanes 16-31 hold rows M=8+j
        if (nloc == 0) {
            const int mbase = (lane >> 4) * 8;
            #pragma unroll
            for (int j = 0; j < 8; ++j) red[w][mbase + j] = p[j];
        }
        __syncthreads();

        // ---- cross-wave reduction over the 8 hidden slices + bias + store ----
        if (tid < GROUP) {
            float acc = b2v;
            #pragma unroll
            for (int wv = 0; wv < NWAVES; ++wv) acc += red[wv][tid];
            out[g * GROUP + tid] = acc;
        }
        __syncthreads();   // protect zt/red before next group's overwrite
    }
}

extern "C" void kernel_launch(void* const* d_in, const int* in_sizes, int n_in,
                              void* d_out, int out_size, void* d_ws, size_t ws_size,
                              hipStream_t stream) {
    const float*     zs  = (const float*)d_in[0];
    const float*     zc  = (const float*)d_in[1];
    const long long* row = (const long long*)d_in[2];
    const long long* col = (const long long*)d_in[3];
    const float*     W1  = (const float*)d_in[4];
    const float*     b1  = (const float*)d_in[5];
    const float*     W2  = (const float*)d_in[6];
    const float*     b2  = (const float*)d_in[7];
    float* out = (float*)d_out;

    const int n_edges = in_sizes[2];
    int ngroups = n_edges / GROUP;
    if (ngroups < 1) ngroups = 1;
    int grid = ngroups < 1024 ? ngroups : 1024;

    hipLaunchKernelGGL(edge_decoder_kernel, dim3(grid), dim3(256), 0, stream,
                       zs, zc, row, col, W1, b1, W2, b2, out, n_edges);
}